// DCNv2_36223754175176
// MI455X (gfx1250) — compile-verified
//
#include <hip/hip_runtime.h>
#include <hip/hip_bf16.h>

// Problem constants (match reference)
#define BB     4
#define CIN    64
#define COUT   64
#define KS     3
#define HH     128
#define WW     128
#define PP     1
#define KKT    9                    // K*K taps
#define HWN    (HH*WW)              // 16384
#define KTOT   (CIN*KKT)            // 576 (GEMM K dim, k fastest within ci)
#define OFF_CH (2*KKT)              // 18 offset channels (mask unused in outputs)
#define NPIX   32                   // pixels per block tile
#define OFF_ELEMS ((size_t)BB*OFF_CH*HWN)   // 1,179,648 floats

typedef __attribute__((ext_vector_type(2))) float v2f;
typedef __attribute__((ext_vector_type(8))) float v8f;

// ---------------------------------------------------------------------------
// Kernel 1: offset conv as implicit-GEMM WMMA.
//   om[:, :18] = cm_w(18x576) @ im2col(offset_in)(576 x pixels) + cm_b
// Block = 128 thr (4 waves), 32-pixel tile. Panel 576x32 fp32 in LDS.
// wave -> (oc tile = wave&1, N subtile = wave>>1). oc rows >= 18 discarded.
// ---------------------------------------------------------------------------
__global__ __launch_bounds__(128)
void offset_conv_wmma_kernel(const float* __restrict__ off_in,  // (B,CIN,H,W)
                             const float* __restrict__ cm_w,    // (27,CIN,3,3)
                             const float* __restrict__ cm_b,    // (27,)
                             float* __restrict__ out_off)       // (B,18,H,W)
{
    __shared__ float sB[KTOT * NPIX];      // 72 KB im2col panel [kk][pixel]

    const int tid   = threadIdx.x;
    const int tile  = blockIdx.x;          // 32-pixel tile in flattened HW
    const int b     = blockIdx.y;
    const int pbase = tile * NPIX;

    // ---- build im2col panel: 576 x 32 (zero padding at borders) ----
    const float* xb = off_in + (size_t)b * CIN * HWN;
    for (int e = tid; e < KTOT * NPIX; e += 128) {
        const int kk = e >> 5;             // ci*9 + (ky*3+kx)
        const int p  = e & (NPIX - 1);
        const int ci = kk / KKT;
        const int k  = kk - ci * KKT;
        const int ky = k / KS, kx = k - ky * KS;
        const int pg = pbase + p;
        const int h  = pg >> 7, w = pg & (WW - 1);
        const int ih = h + ky - PP, iw = w + kx - PP;
        float v = 0.f;
        if (ih >= 0 && ih < HH && iw >= 0 && iw < WW)
            v = xb[(size_t)ci * HWN + ih * WW + iw];
        sB[kk * NPIX + p] = v;
    }
    __syncthreads();

    // ---- WMMA: 144 K-steps of f32 16x16x4 per wave ----
    const int wave = tid >> 5, lane = tid & 31;
    const int ln = lane & 15, half = lane >> 4;
    const int octile = wave & 1;           // 0: oc 0-15, 1: oc 16-17 (+pad)
    const int ns     = wave >> 1;          // N subtile (pixels 0-15 / 16-31)

    const int co_row = octile * 16 + ln;
    const int rowc   = min(co_row, 3 * KKT - 1);   // clamp cm_w row (27 rows)
    const float* wr  = cm_w + (size_t)rowc * KTOT;

    v8f acc = {};
    for (int k0 = 0; k0 < KTOT; k0 += 4) {
        const int kA = k0 + (half << 1);   // lanes 0-15: K=k0,k0+1; 16-31: +2,+3
        v2f a, bf;
        a.x  = wr[kA];
        a.y  = wr[kA + 1];
        bf.x = sB[(kA    ) * NPIX + ns * 16 + ln];
        bf.y = sB[(kA + 1) * NPIX + ns * 16 + ln];
        acc = __builtin_amdgcn_wmma_f32_16x16x4_f32(
            false, a, false, bf, (short)0, acc, false, false);
    }

    // D layout: lane l, vgpr v -> M = v + 8*(l>=16), N = l&15
    #pragma unroll
    for (int v = 0; v < 8; ++v) {
        const int oc = octile * 16 + v + (half << 3);
        if (oc < OFF_CH) {
            out_off[((size_t)b * OFF_CH + oc) * HWN + pbase + ns * 16 + ln] =
                acc[v] + cm_b[oc];
        }
    }
}

// ---------------------------------------------------------------------------
// Kernel 2: fused deformable sampling + WMMA GEMM.
// Block = 128 thr (4 waves), 32-pixel tile.
//   Phase 1: bilinear tables for 32 px x 9 taps
//   Phase 2: 576(K) x 32(px) fp32 sample panel in LDS (shared by 4 waves)
//   Phase 3: wave -> one 16-Cout tile; per K-step: 1 A load, 2 B frags,
//            2 WMMAs (two N subtiles) -> weight L2 traffic halved.
// ---------------------------------------------------------------------------
__global__ __launch_bounds__(128)
void dcn_sample_gemm_kernel(const float* __restrict__ x,      // (B,CIN,H,W)
                            const float* __restrict__ offs,   // (B,18,H,W)
                            const float* __restrict__ weight, // (COUT,KTOT)
                            const float* __restrict__ bias,   // (COUT,)
                            float* __restrict__ y)            // (B,COUT,H,W)
{
    __shared__ float sB[KTOT * NPIX];        // 72 KB sample panel [kk][pixel]
    __shared__ int   sIdx[NPIX * KKT * 4];   // 4.5 KB neighbor indices
    __shared__ float sWt [NPIX * KKT * 4];   // 4.5 KB bilinear weights

    const int tid   = threadIdx.x;
    const int tile  = blockIdx.x;            // 32-pixel tile in flattened HW
    const int b     = blockIdx.y;
    const int pbase = tile * NPIX;

    // ---- Phase 1: bilinear setup (32 pixels x 9 taps = 288 combos) ----
    for (int e = tid; e < NPIX * KKT; e += 128) {
        const int p  = e / KKT;
        const int k  = e - p * KKT;
        const int pg = pbase + p;
        const int h  = pg >> 7;
        const int w  = pg & (WW - 1);

        const float dy = offs[((size_t)b * OFF_CH + 2 * k    ) * HWN + pg];
        const float dx = offs[((size_t)b * OFF_CH + 2 * k + 1) * HWN + pg];

        const float py = dy + (float)(h - PP + k / KS);
        const float px = dx + (float)(w - PP + k % KS);
        const float fy0 = floorf(py), fx0 = floorf(px);
        const float wy = py - fy0,    wx = px - fx0;
        const int iy0 = (int)fy0, ix0 = (int)fx0;
        const int iy1 = iy0 + 1,  ix1 = ix0 + 1;

        const float vy0 = (iy0 >= 0 && iy0 < HH) ? 1.f : 0.f;
        const float vy1 = (iy1 >= 0 && iy1 < HH) ? 1.f : 0.f;
        const float vx0 = (ix0 >= 0 && ix0 < WW) ? 1.f : 0.f;
        const float vx1 = (ix1 >= 0 && ix1 < WW) ? 1.f : 0.f;

        const int cy0 = min(max(iy0, 0), HH - 1);
        const int cy1 = min(max(iy1, 0), HH - 1);
        const int cx0 = min(max(ix0, 0), WW - 1);
        const int cx1 = min(max(ix1, 0), WW - 1);

        const int c4 = e * 4;
        sIdx[c4 + 0] = cy0 * WW + cx0;  sWt[c4 + 0] = (1.f - wy) * (1.f - wx) * vy0 * vx0;
        sIdx[c4 + 1] = cy0 * WW + cx1;  sWt[c4 + 1] = (1.f - wy) * wx         * vy0 * vx1;
        sIdx[c4 + 2] = cy1 * WW + cx0;  sWt[c4 + 2] = wy * (1.f - wx)         * vy1 * vx0;
        sIdx[c4 + 3] = cy1 * WW + cx1;  sWt[c4 + 3] = wy * wx                 * vy1 * vx1;
    }
    __syncthreads();

    // ---- Phase 2: build 576 x 32 sample panel (18432 bilinear gathers) ----
    const float* xb = x + (size_t)b * CIN * HWN;
    for (int e = tid; e < KTOT * NPIX; e += 128) {
        const int kk = e >> 5;             // K index: ci*9 + k
        const int p  = e & (NPIX - 1);     // 32 consecutive threads share (ci,k)
        const int ci = kk / KKT;
        const int k  = kk - ci * KKT;
        const int c4 = ((e & (NPIX - 1)) * KKT + k) * 4;   // (p*9+k)*4
        const float* xc = xb + (size_t)ci * HWN;
        float v = sWt[c4 + 0] * xc[sIdx[c4 + 0]]
                + sWt[c4 + 1] * xc[sIdx[c4 + 1]]
                + sWt[c4 + 2] * xc[sIdx[c4 + 2]]
                + sWt[c4 + 3] * xc[sIdx[c4 + 3]];
        sB[kk * NPIX + p] = v;
    }
    __syncthreads();

    // ---- Phase 3: WMMA. wave -> 16-Cout tile; 1 A load feeds 2 WMMAs ----
    const int wave = tid >> 5, lane = tid & 31;
    const int ln = lane & 15, half = lane >> 4;
    const int co = wave * 16 + ln;
    const float* wr = weight + (size_t)co * KTOT;

    v8f acc0 = {}, acc1 = {};
    for (int k0 = 0; k0 < KTOT; k0 += 4) {
        const int kA = k0 + (half << 1);
        v2f a, b0, b1;
        a.x = wr[kA];
        a.y = wr[kA + 1];
        const float* r0 = &sB[kA * NPIX + ln];
        b0.x = r0[0];          // N subtile 0, K row kA
        b0.y = r0[NPIX];       // N subtile 0, K row kA+1
        b1.x = r0[16];         // N subtile 1, K row kA
        b1.y = r0[NPIX + 16];  // N subtile 1, K row kA+1
        acc0 = __builtin_amdgcn_wmma_f32_16x16x4_f32(
            false, a, false, b0, (short)0, acc0, false, false);
        acc1 = __builtin_amdgcn_wmma_f32_16x16x4_f32(
            false, a, false, b1, (short)0, acc1, false, false);
    }

    // D layout: lane l, vgpr v -> M = v + 8*(l>=16), N = l&15
    float* yb = y + (size_t)b * COUT * HWN + pbase;
    #pragma unroll
    for (int v = 0; v < 8; ++v) {
        const int M  = v + (half << 3);
        const int oc = wave * 16 + M;
        const float bv = bias[oc];
        yb[(size_t)oc * HWN + ln]      = acc0[v] + bv;
        yb[(size_t)oc * HWN + 16 + ln] = acc1[v] + bv;
    }
}

// ---------------------------------------------------------------------------
extern "C" void kernel_launch(void* const* d_in, const int* in_sizes, int n_in,
                              void* d_out, int out_size, void* d_ws, size_t ws_size,
                              hipStream_t stream) {
    const float* x         = (const float*)d_in[0];
    const float* offset_in = (const float*)d_in[1];
    const float* weight    = (const float*)d_in[2];
    const float* bias      = (const float*)d_in[3];
    const float* cm_w      = (const float*)d_in[4];
    const float* cm_b      = (const float*)d_in[5];

    float* out     = (float*)d_out;
    float* out_off = out;               // (B,18,H,W)
    float* out_y   = out + OFF_ELEMS;   // (B,COUT,H,W)

    offset_conv_wmma_kernel<<<dim3(HWN / NPIX, BB), 128, 0, stream>>>(
        offset_in, cm_w, cm_b, out_off);

    dcn_sample_gemm_kernel<<<dim3(HWN / NPIX, BB), 128, 0, stream>>>(
        x, out_off, weight, bias, out_y);
}